// Attention_67448166417198
// MI455X (gfx1250) — compile-verified
//
#include <hip/hip_runtime.h>

typedef __attribute__((ext_vector_type(2))) float v2f;
typedef __attribute__((ext_vector_type(8))) float v8f;
typedef __attribute__((ext_vector_type(4))) int   v4i;
typedef __attribute__((address_space(1))) v4i*    gptr_v4i;  // global int4*
typedef __attribute__((address_space(3))) v4i*    sptr_v4i;  // LDS int4*

#if __has_builtin(__builtin_amdgcn_global_load_async_to_lds_b128) && \
    __has_builtin(__builtin_amdgcn_s_wait_asynccnt)
#define HAVE_ASYNC_LDS 1
#else
#define HAVE_ASYNC_LDS 0
#endif

#define B_   8
#define C_   192
#define H_   128
#define W_   128
#define NH_  6
#define HC_  32        // C_ / NH_
#define HW_  16384     // H_ * W_
#define OC_  576       // 3 * C_
#define BH_  48        // B_ * NH_
#define NSPLIT 16
#define NCHUNK 1024    // HW_ / NSPLIT
#define GEMM_NB 64
#define NSPAN  256     // HW_ / GEMM_NB

// ---------------------------------------------------------------------------
// Kernel 1: grouped 3x3 conv (groups = C, out = 3C), zero padding, + bias.
// JAX grouping: output channel o uses input channel o/3, weight w[o][0][ky][kx].
// One block = one output row (b, o, y); 128 threads = x. 3x130 LDS tile.
// Prefetch the y+2 input row to keep the 3x row re-reads resident in cache.
// ---------------------------------------------------------------------------
__global__ void conv_qkv_kernel(const float* __restrict__ x,
                                const float* __restrict__ w,
                                const float* __restrict__ bias,
                                float* __restrict__ qkv) {
  const int xx = threadIdx.x;        // 0..127
  const int y  = blockIdx.x;         // 0..127
  const int o  = blockIdx.y;         // 0..575
  const int b  = blockIdx.z;         // 0..7
  const int ic = o / 3;
  const float* xin = x + ((size_t)b * C_ + ic) * (size_t)HW_;
  __shared__ float tile[3][W_ + 2];
  #pragma unroll
  for (int r = 0; r < 3; ++r) {
    const int yy = y - 1 + r;
    const float v = (yy >= 0 && yy < H_) ? xin[yy * W_ + xx] : 0.0f;
    tile[r][xx + 1] = v;
    if (xx == 0)      tile[r][0]      = 0.0f;   // left zero-pad
    if (xx == W_ - 1) tile[r][W_ + 1] = 0.0f;   // right zero-pad
  }
  if (y + 2 < H_) __builtin_prefetch(xin + (y + 2) * W_ + xx, 0, 1);
  __syncthreads();
  const float* wp = w + o * 9;
  float acc = bias[o];
  #pragma unroll
  for (int r = 0; r < 3; ++r)
    #pragma unroll
    for (int c = 0; c < 3; ++c)
      acc = fmaf(wp[r * 3 + c], tile[r][xx + c], acc);
  qkv[(((size_t)b * OC_ + o) * H_ + y) * W_ + xx] = acc;
}

// ---------------------------------------------------------------------------
// Kernel 2: per-(b,h) Gram partials S += Q*K^T over an N-chunk of 1024, via
// V_WMMA_F32_16X16X4_F32, plus squared-norm partials for q and k rows.
// grid (48, 16); block = 128 threads = 4 waves; wave w -> 16x16 tile (w>>1, w&1).
// Deterministic: partials land in distinct slots, reduced later in fixed order.
// ---------------------------------------------------------------------------
__global__ void gram_kernel(const float* __restrict__ qkv,
                            float* __restrict__ Spart,
                            float* __restrict__ Npart) {
  const int bh    = blockIdx.x;        // 0..47
  const int split = blockIdx.y;        // 0..15
  const int b = bh / NH_, h = bh % NH_;
  const int lane  = threadIdx.x & 31;
  const int wave  = threadIdx.x >> 5;  // 0..3
  const int ti = wave >> 1, tj = wave & 1;
  const int r     = lane & 15;         // A row / B col / D col
  const int khalf = lane >> 4;         // K sub-pair selector
  const size_t qbase = ((size_t)b * OC_ + h * HC_) * HW_;          // q channels
  const size_t kbase = ((size_t)b * OC_ + C_ + h * HC_) * HW_;     // k channels
  const float* qrow = qkv + qbase + (size_t)(ti * 16 + r) * HW_;
  const float* krow = qkv + kbase + (size_t)(tj * 16 + r) * HW_;
  const int n0 = split * NCHUNK;

  v8f acc = {0.f, 0.f, 0.f, 0.f, 0.f, 0.f, 0.f, 0.f};
  for (int n = n0; n < n0 + NCHUNK; n += 4) {
    const int nn = n + 2 * khalf;
    v2f a  = *(const v2f*)(qrow + nn);   // A[row, k..k+1] (contiguous in N)
    v2f bb = *(const v2f*)(krow + nn);   // B[k..k+1, col] = K[col, n..n+1]
    acc = __builtin_amdgcn_wmma_f32_16x16x4_f32(
        false, a, false, bb, (short)0, acc, false, false);
  }
  float* sp = Spart + ((size_t)bh * NSPLIT + split) * (HC_ * HC_);
  #pragma unroll
  for (int i = 0; i < 8; ++i) {
    const int rr = ti * 16 + i + khalf * 8;      // D: M = i + 8*(lane>>4)
    sp[rr * HC_ + tj * 16 + r] = acc[i];
  }

  // squared-norm partials: 128 threads -> (tensor, half, row)
  const int tensor = threadIdx.x >> 6;        // 0=q, 1=k
  const int half   = (threadIdx.x >> 5) & 1;  // which 512-chunk
  const int rowc   = threadIdx.x & 31;
  const float4* src4 = (const float4*)(qkv + (tensor ? kbase : qbase) +
                                       (size_t)rowc * HW_ + n0 + half * (NCHUNK / 2));
  float s = 0.f;
  #pragma unroll 4
  for (int i = 0; i < NCHUNK / 8; ++i) {
    const float4 v = src4[i];
    s = fmaf(v.x, v.x, s); s = fmaf(v.y, v.y, s);
    s = fmaf(v.z, v.z, s); s = fmaf(v.w, v.w, s);
  }
  Npart[(((size_t)bh * NSPLIT + split) * 4 + tensor * 2 + half) * HC_ + rowc] = s;
}

// ---------------------------------------------------------------------------
// Kernel 3: reduce partials, normalize Gram by row norms (== normalizing q,k
// before the dot product), scale by temperature, safe softmax over last dim.
// grid 48; 32 threads; thread c owns attn row c.
// ---------------------------------------------------------------------------
__global__ void softmax_kernel(const float* __restrict__ Spart,
                               const float* __restrict__ Npart,
                               const float* __restrict__ temperature,
                               float* __restrict__ attn) {
  const int bh = blockIdx.x;
  const int h  = bh % NH_;
  const int c  = threadIdx.x;   // 0..31
  float qn2 = 0.f, kn2 = 0.f;
  for (int s = 0; s < NSPLIT; ++s) {
    const float* np = Npart + ((size_t)bh * NSPLIT + s) * (4 * HC_);
    qn2 += np[c] + np[HC_ + c];
    kn2 += np[2 * HC_ + c] + np[3 * HC_ + c];
  }
  __shared__ float qinv[HC_], kinv[HC_];
  const float eps = 1e-12f;
  qinv[c] = 1.0f / fmaxf(sqrtf(qn2), eps);
  kinv[c] = 1.0f / fmaxf(sqrtf(kn2), eps);
  __syncthreads();

  float row[HC_];
  #pragma unroll
  for (int d = 0; d < HC_; ++d) row[d] = 0.f;
  for (int s = 0; s < NSPLIT; ++s) {
    const float* sp = Spart + ((size_t)bh * NSPLIT + s) * (HC_ * HC_) + c * HC_;
    #pragma unroll
    for (int d = 0; d < HC_; ++d) row[d] += sp[d];
  }
  const float t = temperature[h];
  float m = -3.4e38f;
  #pragma unroll
  for (int d = 0; d < HC_; ++d) {
    row[d] = row[d] * qinv[c] * kinv[d] * t;
    m = fmaxf(m, row[d]);
  }
  float sum = 0.f;
  #pragma unroll
  for (int d = 0; d < HC_; ++d) { row[d] = expf(row[d] - m); sum += row[d]; }
  const float inv = 1.0f / sum;
  float* ap = attn + (size_t)bh * (HC_ * HC_) + c * HC_;
  #pragma unroll
  for (int d = 0; d < HC_; ++d) ap[d] = row[d] * inv;
}

// ---------------------------------------------------------------------------
// Kernel 4: fold 1x1 conv into attention: F[b, o, h*32+d] =
//   sum_c w_out[o, h*32+c] * attn[b,h,c,d].  grid (8,6); block (32,8).
// ---------------------------------------------------------------------------
__global__ void fuse_wout_kernel(const float* __restrict__ w_out,
                                 const float* __restrict__ attn,
                                 float* __restrict__ F) {
  const int b = blockIdx.x, h = blockIdx.y;
  __shared__ float A[HC_ * HC_];
  const int tid = threadIdx.y * 32 + threadIdx.x;
  for (int i = tid; i < HC_ * HC_; i += 256)
    A[i] = attn[(size_t)(b * NH_ + h) * (HC_ * HC_) + i];
  __syncthreads();
  const int d = threadIdx.x;
  for (int o = threadIdx.y; o < C_; o += 8) {
    const float* wrow = w_out + (size_t)o * C_ + h * HC_;
    float s = 0.f;
    #pragma unroll
    for (int cc = 0; cc < HC_; ++cc) s = fmaf(wrow[cc], A[cc * HC_ + d], s);
    F[((size_t)b * C_ + o) * C_ + h * HC_ + d] = s;
  }
}

// ---------------------------------------------------------------------------
// Kernel 5: out[b] = F_b (192x192) @ V_b (192x16384) + b_out, WMMA f32.
// grid (64, 8); block 384 threads = 12 waves; wave w owns output rows
// [16w, 16w+16) and keeps its 16x192 slice of F resident as 48 A fragments.
// V tiles (192x16 = 12KB) are double-buffered in LDS; the next tile streams
// in via GLOBAL_LOAD_ASYNC_TO_LDS_B128 (ASYNCcnt path, no VGPR staging)
// while the current tile's 48 WMMAs execute.
// ---------------------------------------------------------------------------
__global__ void gemm_out_kernel(const float* __restrict__ F,
                                const float* __restrict__ qkv,
                                const float* __restrict__ bias,
                                float* __restrict__ out) {
  const int b     = blockIdx.y;
  const int nbase = blockIdx.x * NSPAN;
  const int wave  = threadIdx.x >> 5;   // 0..11
  const int lane  = threadIdx.x & 31;
  const int row   = lane & 15;
  const int khalf = lane >> 4;
  const float* Fb = F + (size_t)b * C_ * C_;
  const float* Vb = qkv + ((size_t)b * OC_ + 2 * C_) * HW_;   // v channels

  // Preload A fragments: A[row, k..k+1] with k = 4*kk + 2*khalf (row-major F).
  v2f afrag[48];
  const float* arow = Fb + (size_t)(wave * 16 + row) * C_;
  #pragma unroll
  for (int kk = 0; kk < 48; ++kk)
    afrag[kk] = *(const v2f*)(arow + kk * 4 + khalf * 2);

  __shared__ float bs[2][C_ * 16];   // double-buffered V[0:192, n0:n0+16]
  const int NT = NSPAN / 16;

#if HAVE_ASYNC_LDS
  // wave w async-copies rows [16w, 16w+16): 2 x b128 (8 rows x 64B each).
  const int akr  = lane >> 2;          // row within 8-row group
  const int acol = (lane & 3) * 4;     // float column offset (16B granule)
  auto issue_fill = [&](int buf, int n0) {
    #pragma unroll
    for (int j = 0; j < 2; ++j) {
      const int kr = wave * 16 + j * 8 + akr;
      __builtin_amdgcn_global_load_async_to_lds_b128(
          (gptr_v4i)(Vb + (size_t)kr * HW_ + n0 + acol),
          (sptr_v4i)(&bs[buf][kr * 16 + acol]), 0, 0);
    }
  };
  issue_fill(0, nbase);
  __builtin_amdgcn_s_wait_asynccnt(0);
  __syncthreads();
#else
  auto plain_fill = [&](int buf, int n0) {
    for (int i = threadIdx.x; i < C_ * 16; i += 384) {
      const int kr = i >> 4, col = i & 15;
      bs[buf][i] = Vb[(size_t)kr * HW_ + n0 + col];
    }
  };
  plain_fill(0, nbase);
  __syncthreads();
#endif

  for (int nt = 0; nt < NT; ++nt) {
    const int n0  = nbase + nt * 16;
    const int cur = nt & 1;
#if HAVE_ASYNC_LDS
    if (nt + 1 < NT) issue_fill(cur ^ 1, n0 + 16);   // overlap with compute
#else
    if (nt + 1 < NT) plain_fill(cur ^ 1, n0 + 16);
#endif

    v8f acc = {0.f, 0.f, 0.f, 0.f, 0.f, 0.f, 0.f, 0.f};
    #pragma unroll
    for (int kk = 0; kk < 48; ++kk) {
      const int kr = kk * 4 + khalf * 2;
      v2f bf;
      bf.x = bs[cur][kr * 16 + row];
      bf.y = bs[cur][(kr + 1) * 16 + row];
      acc = __builtin_amdgcn_wmma_f32_16x16x4_f32(
          false, afrag[kk], false, bf, (short)0, acc, false, false);
    }
    // D layout: VGPR i -> M = i + 8*khalf, N = lane&15
    #pragma unroll
    for (int i = 0; i < 8; ++i) {
      const int ch = wave * 16 + khalf * 8 + i;
      out[((size_t)b * C_ + ch) * HW_ + n0 + row] = acc[i] + bias[ch];
    }
#if HAVE_ASYNC_LDS
    __builtin_amdgcn_s_wait_asynccnt(0);
#endif
    __syncthreads();
  }
}

// ---------------------------------------------------------------------------
extern "C" void kernel_launch(void* const* d_in, const int* in_sizes, int n_in,
                              void* d_out, int out_size, void* d_ws, size_t ws_size,
                              hipStream_t stream) {
  (void)in_sizes; (void)n_in; (void)out_size; (void)ws_size;
  const float* x     = (const float*)d_in[0];
  const float* w_qkv = (const float*)d_in[1];
  const float* b_qkv = (const float*)d_in[2];
  const float* temp  = (const float*)d_in[3];
  const float* w_out = (const float*)d_in[4];
  const float* b_out = (const float*)d_in[5];
  float* out = (float*)d_out;

  // workspace layout (floats): qkv | Spart | Npart | attn | F  (~307 MB)
  float* qkv   = (float*)d_ws;
  float* Spart = qkv   + (size_t)B_ * OC_ * HW_;
  float* Npart = Spart + (size_t)BH_ * NSPLIT * HC_ * HC_;
  float* attn  = Npart + (size_t)BH_ * NSPLIT * 4 * HC_;
  float* F     = attn  + (size_t)BH_ * HC_ * HC_;

  conv_qkv_kernel<<<dim3(H_, OC_, B_), dim3(W_), 0, stream>>>(x, w_qkv, b_qkv, qkv);
  gram_kernel<<<dim3(BH_, NSPLIT), dim3(128), 0, stream>>>(qkv, Spart, Npart);
  softmax_kernel<<<dim3(BH_), dim3(32), 0, stream>>>(Spart, Npart, temp, attn);
  fuse_wout_kernel<<<dim3(B_, NH_), dim3(32, 8), 0, stream>>>(w_out, attn, F);
  gemm_out_kernel<<<dim3(GEMM_NB, B_), dim3(384), 0, stream>>>(F, qkv, b_out, out);
}